// Q_PGA_Unfold_67250597921283
// MI455X (gfx1250) — compile-verified
//
#include <hip/hip_runtime.h>
#include <hip/hip_bf16.h>

// ---------------- problem constants ----------------
#define KUSR  4
#define NTX   64
#define NRFX  4
#define MXX   4
#define NQX   8
#define BX    256
#define NOUTX 20
#define NINX  20
#define OMEGAX 0.5f
#define INV_LN2_X2 2.8853901817f   // 2/ln2
#define TAU_GSCL   4.8828125e-4f   // 2/(64*64)

typedef float v2f __attribute__((ext_vector_type(2)));
typedef float v8f __attribute__((ext_vector_type(8)));

struct cplx { float x, y; };
__device__ __forceinline__ cplx cmk(float a, float b){ cplx c; c.x=a; c.y=b; return c; }
__device__ __forceinline__ cplx cadd(cplx a, cplx b){ return cmk(a.x+b.x, a.y+b.y); }
__device__ __forceinline__ cplx csub(cplx a, cplx b){ return cmk(a.x-b.x, a.y-b.y); }
__device__ __forceinline__ cplx cmul(cplx a, cplx b){ return cmk(a.x*b.x-a.y*b.y, a.x*b.y+a.y*b.x); }
__device__ __forceinline__ cplx cmulc(cplx a, cplx b){ /* a*conj(b) */ return cmk(a.x*b.x+a.y*b.y, a.y*b.x-a.x*b.y); }

// ---- f32 WMMA (16x16x4, exact f32 accumulate; A=16x4, B=4x16, C/D=16x16) ----
__device__ __forceinline__ v8f wmma4(v2f a, v2f b, v8f c){
  return __builtin_amdgcn_wmma_f32_16x16x4_f32(false, a, false, b, (short)0, c, false, false);
}

// ============= 4x4 complex Hermitian helpers (single-thread) =============
__device__ __forceinline__ void buildG(const float* Are, const float* Aim, int k, cplx G[4][4]){
  for (int m=0;m<4;m++) for (int n=0;n<4;n++){
    cplx acc = cmk((m==n)?1.f:0.f, 0.f);
    for (int s=0;s<4;s++){
      cplx am = cmk(Are[k*16+m*4+s], Aim[k*16+m*4+s]);
      cplx an = cmk(Are[k*16+n*4+s], Aim[k*16+n*4+s]);
      acc = cadd(acc, cmulc(am, an));
    }
    G[m][n] = acc;
  }
}
__device__ __forceinline__ float chol4(const cplx G[4][4], cplx L[4][4]){
  float ld = 0.f;
  for (int i=0;i<4;i++){
    float d = G[i][i].x;
    for (int p=0;p<i;p++) d -= L[i][p].x*L[i][p].x + L[i][p].y*L[i][p].y;
    d = fmaxf(d, 1e-20f);
    float lii = sqrtf(d); ld += logf(lii);
    L[i][i] = cmk(lii, 0.f);
    float inv = 1.f/lii;
    for (int j=i+1;j<4;j++){
      cplx acc = G[j][i];
      for (int p=0;p<i;p++) acc = csub(acc, cmulc(L[j][p], L[i][p]));
      L[j][i] = cmk(acc.x*inv, acc.y*inv);
    }
    for (int j=0;j<i;j++) L[j][i] = cmk(0.f,0.f);
  }
  return 2.f*ld;
}
__device__ __forceinline__ void ginv4(const cplx L[4][4], cplx Gi[4][4]){
  cplx Li[4][4];
  for (int i=0;i<4;i++) for (int j=0;j<4;j++) Li[i][j]=cmk(0.f,0.f);
  for (int i=0;i<4;i++){
    Li[i][i] = cmk(1.f/L[i][i].x, 0.f);
    for (int j=i+1;j<4;j++){
      cplx acc = cmk(0.f,0.f);
      for (int p=i;p<j;p++) acc = cadd(acc, cmul(L[j][p], Li[p][i]));
      float inv = 1.f/L[j][j].x;
      Li[j][i] = cmk(-acc.x*inv, -acc.y*inv);
    }
  }
  for (int m=0;m<4;m++) for (int n=0;n<4;n++){
    cplx acc = cmk(0.f,0.f);
    for (int p=0;p<4;p++) acc = cadd(acc, cmulc(Li[p][n], Li[p][m])); // conj(Li[p][m])*Li[p][n]
    Gi[m][n] = cmk(acc.x, -acc.y); // (above computed conj) -> fix orientation
  }
}

// ============= 8-qubit statevector on one wave32 (8 amps/lane) =============
__device__ __forceinline__ void apply_gate(cplx a[8], int w, cplx u00, cplx u01, cplx u10, cplx u11, int lane){
  int p = 7 - w;
  if (p >= 3){
    int lm = 1 << (p-3);
    int bit = (lane >> (p-3)) & 1;
    #pragma unroll
    for (int j=0;j<8;j++){
      cplx bb; bb.x = __shfl_xor(a[j].x, lm, 32); bb.y = __shfl_xor(a[j].y, lm, 32);
      cplx x0 = bit ? bb : a[j];
      cplx x1 = bit ? a[j] : bb;
      cplx n0 = cadd(cmul(u00,x0), cmul(u01,x1));
      cplx n1 = cadd(cmul(u10,x0), cmul(u11,x1));
      a[j] = bit ? n1 : n0;
    }
  } else {
    int m = 1 << p;
    #pragma unroll
    for (int j=0;j<8;j++){
      if ((j & m) == 0){
        int j1 = j | m;
        cplx x0 = a[j], x1 = a[j1];
        a[j]  = cadd(cmul(u00,x0), cmul(u01,x1));
        a[j1] = cadd(cmul(u10,x0), cmul(u11,x1));
      }
    }
  }
}
__device__ __forceinline__ void apply_cnot(cplx a[8], int c, int t, int lane){
  int pc = 7 - c, pt = 7 - t;
  if (pt >= 3){
    int lm = 1 << (pt-3);
    #pragma unroll
    for (int j=0;j<8;j++){
      cplx bb; bb.x = __shfl_xor(a[j].x, lm, 32); bb.y = __shfl_xor(a[j].y, lm, 32);
      int cb = (pc >= 3) ? ((lane >> (pc-3)) & 1) : ((j >> pc) & 1);
      if (cb) a[j] = bb;
    }
  } else {
    int m = 1 << pt;
    #pragma unroll
    for (int j=0;j<8;j++){
      if ((j & m) == 0){
        int j1 = j | m;
        int cb = (pc >= 3) ? ((lane >> (pc-3)) & 1) : ((j >> pc) & 1);
        if (cb){ cplx tmp=a[j]; a[j]=a[j1]; a[j1]=tmp; }
      }
    }
  }
}
__device__ void run_circuit(const float* ang, const float* qw, float* eout){
  int lane = threadIdx.x & 31;
  cplx a[8];
  #pragma unroll
  for (int j=0;j<8;j++) a[j] = cmk(0.f,0.f);
  if (lane == 0) a[0].x = 1.f;
  // angle encoding: RY(t) then RZ(t) per wire
  for (int w=0; w<8; ++w){
    float s, c; __sincosf(0.5f*ang[w], &s, &c);
    apply_gate(a, w, cmk(c,0.f), cmk(-s,0.f), cmk(s,0.f), cmk(c,0.f), lane);
    apply_gate(a, w, cmk(c,-s), cmk(0.f,0.f), cmk(0.f,0.f), cmk(c,s), lane);
  }
  // 3 entangling layers: Rot + CNOT ring
  for (int l=0;l<3;l++){
    for (int w=0;w<8;w++){
      float phi = qw[(l*8+w)*3+0], th = qw[(l*8+w)*3+1], om = qw[(l*8+w)*3+2];
      float st_, ct; __sincosf(0.5f*th, &st_, &ct);
      float sa, ca; __sincosf(0.5f*(phi+om), &sa, &ca);
      float sb, cb; __sincosf(0.5f*(phi-om), &sb, &cb);
      cplx u00 = cmk( ct*ca, -ct*sa);
      cplx u01 = cmk(-st_*cb, -st_*sb);
      cplx u10 = cmk( st_*cb, -st_*sb);
      cplx u11 = cmk( ct*ca,  ct*sa);
      apply_gate(a, w, u00,u01,u10,u11, lane);
    }
    for (int i=0;i<7;i++) apply_cnot(a, i, i+1, lane);
    apply_cnot(a, 7, 0, lane);
  }
  // <Z_w> expectation values (wave-reduced, uniform)
  #pragma unroll
  for (int w=0;w<8;w++){
    int p = 7 - w;
    float s = 0.f;
    #pragma unroll
    for (int j=0;j<8;j++){
      int bit = (p>=3) ? ((lane>>(p-3))&1) : ((j>>p)&1);
      float pr = a[j].x*a[j].x + a[j].y*a[j].y;
      s += bit ? -pr : pr;
    }
    for (int off=16; off; off>>=1) s += __shfl_xor(s, off, 32);
    eout[w] = s;
  }
}

// ============================ kernels ============================

// F = F0/|F0|; W = W0; FW = F@W; W,FW *= sqrt(Pt/||FW||^2)
__global__ __launch_bounds__(256) void k_init(
    const float* __restrict__ F0re, const float* __restrict__ F0im,
    const float* __restrict__ W0re, const float* __restrict__ W0im,
    const int* __restrict__ PtPtr, float* Fp, float* Wp, float* FWp)
{
  int b = blockIdx.x, tid = threadIdx.x;
  __shared__ float red[256];
  {
    float re = F0re[b*256 + tid], im = F0im[b*256 + tid];
    float inv = 1.f/(sqrtf(re*re+im*im) + 1e-12f);
    Fp[(b*256+tid)*2]   = re*inv;
    Fp[(b*256+tid)*2+1] = im*inv;
  }
  if (tid < 64){
    Wp[(b*64+tid)*2]   = W0re[b*64+tid];
    Wp[(b*64+tid)*2+1] = W0im[b*64+tid];
  }
  __syncthreads();
  float acc = 0.f;
  for (int i = tid; i < 1024; i += 256){
    int k = i >> 8, t = (i >> 2) & 63, m = i & 3;
    cplx s = cmk(0.f,0.f);
    for (int r=0;r<4;r++){
      cplx f = cmk(Fp[(b*256 + t*4 + r)*2], Fp[(b*256 + t*4 + r)*2+1]);
      cplx w = cmk(Wp[(b*64 + (k*4+r)*4 + m)*2], Wp[(b*64 + (k*4+r)*4 + m)*2+1]);
      s = cadd(s, cmul(f,w));
    }
    int idx = b*1024 + i;
    FWp[idx*2] = s.x; FWp[idx*2+1] = s.y;
    acc += s.x*s.x + s.y*s.y;
  }
  red[tid] = acc; __syncthreads();
  for (int st=128; st; st>>=1){ if (tid<st) red[tid]+=red[tid+st]; __syncthreads(); }
  float scale = sqrtf((float)PtPtr[0] / (red[0] + 1e-12f));
  if (tid < 64){ Wp[(b*64+tid)*2] *= scale; Wp[(b*64+tid)*2+1] *= scale; }
  for (int i = tid; i < 1024; i += 256){
    int idx = b*1024 + i;
    FWp[idx*2] *= scale; FWp[idx*2+1] *= scale;
  }
}

// rate_b = sum_k logdet(I+AA^H)/ln2 ; tau_b = mean|X X^H - R|^2  (X = FW, WMMA tiles)
__global__ __launch_bounds__(512) void k_rate_tau(
    const float* __restrict__ FWp,
    const float* __restrict__ Hre, const float* __restrict__ Him,
    const float* __restrict__ Rre, const float* __restrict__ Rim,
    float* rates, float* taus, int col)
{
  int b = blockIdx.x, tid = threadIdx.x;
  int lane = tid & 31, wave = tid >> 5, hi = lane >> 4, lo = lane & 15;
  __shared__ float Xre[1024], Xim[1024];
  __shared__ float red[512];
  __shared__ float Are[64], Aim[64];
  __shared__ float kout[4];
  for (int i = tid; i < 1024; i += 512){
    int t = i >> 4, mu = i & 15; int k = mu >> 2, m = mu & 3;
    int idx = b*1024 + k*256 + t*4 + m;
    Xre[i] = FWp[idx*2]; Xim[i] = FWp[idx*2+1];
  }
  __syncthreads();
  int row0 = (wave >> 2) * 16, col0 = (wave & 3) * 16;
  v8f tre = {}, tim = {};
  #pragma unroll
  for (int kk=0; kk<4; ++kk){
    int ca = 4*kk + 2*hi;
    v2f aRe, aIm, bRe, bIm, nIm;
    aRe[0]=Xre[(row0+lo)*16+ca]; aRe[1]=Xre[(row0+lo)*16+ca+1];
    aIm[0]=Xim[(row0+lo)*16+ca]; aIm[1]=Xim[(row0+lo)*16+ca+1];
    bRe[0]=Xre[(col0+lo)*16+ca]; bRe[1]=Xre[(col0+lo)*16+ca+1];
    bIm[0]=Xim[(col0+lo)*16+ca]; bIm[1]=Xim[(col0+lo)*16+ca+1];
    nIm[0]=-bIm[0]; nIm[1]=-bIm[1];
    tre = wmma4(aRe,bRe,tre); tre = wmma4(aIm,bIm,tre);   // Re: XreXre^T + XimXim^T
    tim = wmma4(aIm,bRe,tim); tim = wmma4(aRe,nIm,tim);   // Im: XimXre^T - XreXim^T
  }
  float part = 0.f;
  #pragma unroll
  for (int r=0;r<8;r++){
    int m = row0 + r + 8*hi, n = col0 + lo;
    float er = tre[r] - Rre[(b*64+m)*64+n];
    float ei = tim[r] - Rim[(b*64+m)*64+n];
    part += er*er + ei*ei;
  }
  red[tid] = part; __syncthreads();
  for (int st=256; st; st>>=1){ if (tid<st) red[tid]+=red[tid+st]; __syncthreads(); }
  if (tid < 64){
    int k = tid >> 4, m = (tid >> 2) & 3, s = tid & 3;
    const float* hr = Hre + ((size_t)(k*BX + b)*MXX + m)*NTX;
    const float* hii = Him + ((size_t)(k*BX + b)*MXX + m)*NTX;
    cplx a = cmk(0.f,0.f);
    for (int t=0;t<64;t++)
      a = cadd(a, cmul(cmk(hr[t],hii[t]), cmk(Xre[t*16+k*4+s], Xim[t*16+k*4+s])));
    Are[tid] = a.x; Aim[tid] = a.y;
  }
  __syncthreads();
  if (tid < 4){
    cplx G[4][4], L[4][4];
    buildG(Are, Aim, tid, G);
    kout[tid] = chol4(G, L);
  }
  __syncthreads();
  if (tid == 0){
    rates[b*(NOUTX+1)+col] = (kout[0]+kout[1]+kout[2]+kout[3]) * 1.4426950408f;
    taus [b*(NOUTX+1)+col] = red[0] * (1.f/4096.f);
  }
}

// state features -> 80-d projected vector (batch-invariant: computed once)
__global__ __launch_bounds__(512) void k_state(
    const float* __restrict__ Fp, const float* __restrict__ Wp,
    const float* __restrict__ Hre, const float* __restrict__ Him,
    const float* __restrict__ Rre, const float* __restrict__ Rim,
    const float* __restrict__ PW, const float* __restrict__ Pb, float* vout)
{
  int tid = threadIdx.x;
  __shared__ float sbuf[512];
  __shared__ float sfeat[144];
  float accF[8] = {0,0,0,0,0,0,0,0};
  for (int i = tid; i < BX*NTX; i += 512){
    #pragma unroll
    for (int r=0;r<4;r++){
      float re = Fp[(i*4+r)*2], im = Fp[(i*4+r)*2+1];
      accF[r]   += sqrtf(re*re+im*im);
      accF[4+r] += atan2f(im, re);
    }
  }
  for (int q=0;q<8;q++){
    sbuf[tid]=accF[q]; __syncthreads();
    for (int st=256; st; st>>=1){ if (tid<st) sbuf[tid]+=sbuf[tid+st]; __syncthreads(); }
    if (tid==0) sfeat[q] = sbuf[0]*(1.f/16384.f);
    __syncthreads();
  }
  float accW[8] = {0,0,0,0,0,0,0,0};
  for (int i = tid; i < BX*KUSR*NRFX; i += 512){
    #pragma unroll
    for (int m=0;m<4;m++){
      float re = Wp[(i*4+m)*2], im = Wp[(i*4+m)*2+1];
      accW[m]   += sqrtf(re*re+im*im);
      accW[4+m] += atan2f(im, re);
    }
  }
  for (int q=0;q<8;q++){
    sbuf[tid]=accW[q]; __syncthreads();
    for (int st=256; st; st>>=1){ if (tid<st) sbuf[tid]+=sbuf[tid+st]; __syncthreads(); }
    if (tid==0) sfeat[8+q] = sbuf[0]*(1.f/4096.f);
    __syncthreads();
  }
  int tsub = tid & 63, part = tid >> 6;
  float ah = 0.f;
  for (int g = part; g < 4096; g += 8){
    float re = Hre[g*64+tsub], im = Him[g*64+tsub];
    ah += sqrtf(re*re+im*im);
  }
  sbuf[tid]=ah; __syncthreads();
  if (tid < 64){
    float s=0.f; for (int p=0;p<8;p++) s += sbuf[tid+64*p];
    sfeat[16+tid] = s*(1.f/4096.f);
  }
  __syncthreads();
  float ar = 0.f;
  for (int g = part; g < 16384; g += 8){
    float re = Rre[g*64+tsub], im = Rim[g*64+tsub];
    ar += sqrtf(re*re+im*im);
  }
  sbuf[tid]=ar; __syncthreads();
  if (tid < 64){
    float s=0.f; for (int p=0;p<8;p++) s += sbuf[tid+64*p];
    sfeat[80+tid] = s*(1.f/16384.f);
  }
  __syncthreads();
  if (tid < 80){
    float a = Pb[tid];
    for (int j=0;j<144;j++) a += sfeat[j]*PW[j*80+tid];
    vout[tid] = a;
  }
}

// MLP + 8-qubit circuits -> dq (blocks 0..19 = F-blocks, block 20 = W-block)
__global__ __launch_bounds__(64) void k_qblock(
    const float* __restrict__ v,
    const float* FbW1, const float* Fbb1, const float* FbW2, const float* Fbb2,
    const float* Fbqw, const float* FbW3, const float* Fbb3,
    const float* WbW1, const float* Wbb1, const float* WbW2, const float* Wbb2,
    const float* Wbqw, const float* WbW3, const float* Wbb3,
    float* dqF, float* dqW)
{
  int j = blockIdx.x, tid = threadIdx.x, lane = tid & 31, wave = tid >> 5;
  bool isW = (j == NINX);
  const float* W1 = isW ? WbW1 : FbW1 + j*80*32;
  const float* b1 = isW ? Wbb1 : Fbb1 + j*32;
  const float* W2 = isW ? WbW2 : FbW2 + j*32*16;
  const float* b2 = isW ? Wbb2 : Fbb2 + j*16;
  const float* qw = isW ? Wbqw : Fbqw + j*72;
  const float* W3 = isW ? WbW3 : FbW3 + j*NQX*512;
  const float* b3 = isW ? Wbb3 : Fbb3 + j*512;
  int outdim = isW ? 128 : 512;
  float* out = isW ? dqW : dqF + j*512;
  __shared__ float h[32], z[16], e0[8], e1[8], q[8];
  if (tid < 32){
    float a = b1[tid];
    for (int aa=0; aa<80; ++aa) a += v[aa]*W1[aa*32+tid];
    h[tid] = tanhf(a);
  }
  __syncthreads();
  if (tid < 16){
    float a = b2[tid];
    for (int i=0;i<32;i++) a += h[i]*W2[i*16+tid];
    z[tid] = a;
  }
  __syncthreads();
  float ev[8];
  run_circuit(&z[wave*8], qw, ev);
  if (lane == 0){
    float* e = wave ? e1 : e0;
    for (int w=0;w<8;w++) e[w] = ev[w];
  }
  __syncthreads();
  if (tid < 8) q[tid] = 0.5f*(e0[tid]+e1[tid]);
  __syncthreads();
  for (int o = tid; o < outdim; o += 64){
    float a = b3[o];
    for (int i=0;i<8;i++) a += q[i]*W3[i*outdim+o];
    out[o] = a * 5e-4f;
  }
}

// analytic Wirtinger gradients: gFc/gFr (computeW=0) or gWc/gWr (computeW=1)
__global__ __launch_bounds__(512) void k_grads(
    const float* __restrict__ FWp, const float* __restrict__ Fp, const float* __restrict__ Wp,
    const float* __restrict__ Hre, const float* __restrict__ Him,
    const float* __restrict__ Rre, const float* __restrict__ Rim,
    float* gFc, float* gFr, float* gWc, float* gWr, int computeW)
{
  int b = blockIdx.x, tid = threadIdx.x;
  int lane = tid & 31, wave = tid >> 5, hi = lane >> 4, lo = lane & 15;
  __shared__ float Xre[1024], Xim[1024];
  __shared__ float Ere[4096], Eim[4096];
  __shared__ float gXre[1024], gXim[1024];
  __shared__ float Are[64], Aim[64], gAre[64], gAim[64], Pre[64], Pim[64], HFre[64], HFim[64];

  __builtin_prefetch(Hre + (size_t)b*MXX*NTX, 0, 1);
  __builtin_prefetch(Him + (size_t)b*MXX*NTX, 0, 1);

  for (int i = tid; i < 1024; i += 512){
    int t = i >> 4, mu = i & 15; int k = mu >> 2, m = mu & 3;
    int idx = b*1024 + k*256 + t*4 + m;
    Xre[i] = FWp[idx*2]; Xim[i] = FWp[idx*2+1];
  }
  __syncthreads();

  // ---- phase A: all 16 waves compute one 16x16 tile of E = X X^H - R (WMMA) ----
  {
    int row0 = (wave >> 2)*16, col0 = (wave & 3)*16;
    v8f tre = {}, tim = {};
    #pragma unroll
    for (int kk=0; kk<4; ++kk){
      int ca = 4*kk + 2*hi;
      v2f aRe, aIm, bRe, bIm, nIm;
      aRe[0]=Xre[(row0+lo)*16+ca]; aRe[1]=Xre[(row0+lo)*16+ca+1];
      aIm[0]=Xim[(row0+lo)*16+ca]; aIm[1]=Xim[(row0+lo)*16+ca+1];
      bRe[0]=Xre[(col0+lo)*16+ca]; bRe[1]=Xre[(col0+lo)*16+ca+1];
      bIm[0]=Xim[(col0+lo)*16+ca]; bIm[1]=Xim[(col0+lo)*16+ca+1];
      nIm[0]=-bIm[0]; nIm[1]=-bIm[1];
      tre = wmma4(aRe,bRe,tre); tre = wmma4(aIm,bIm,tre);
      tim = wmma4(aIm,bRe,tim); tim = wmma4(aRe,nIm,tim);
    }
    #pragma unroll
    for (int r=0;r<8;r++){
      int m = row0 + r + 8*hi, n = col0 + lo;
      Ere[m*64+n] = tre[r] - Rre[(b*64+m)*64+n];
      Eim[m*64+n] = tim[r] - Rim[(b*64+m)*64+n];
    }
  }
  __syncthreads();

  // ---- phase B: waves 0-3 do gX = (E^T + conj(E)) X via WMMA; waves 8-11 do A / HF ----
  if (wave < 4){
    int row0 = wave*16;
    v8f gre = {}, gim = {};
    for (int kk=0; kk<16; ++kk){
      int c0 = 4*kk + 2*hi;
      int t = row0 + lo;
      v2f mRe, mIm, nIm, bRe, bIm;
      mRe[0] = Ere[c0*64+t]     + Ere[t*64+c0];
      mRe[1] = Ere[(c0+1)*64+t] + Ere[t*64+c0+1];
      mIm[0] = Eim[c0*64+t]     - Eim[t*64+c0];
      mIm[1] = Eim[(c0+1)*64+t] - Eim[t*64+c0+1];
      nIm[0] = -mIm[0]; nIm[1] = -mIm[1];
      bRe[0] = Xre[c0*16+lo]; bRe[1] = Xre[(c0+1)*16+lo];
      bIm[0] = Xim[c0*16+lo]; bIm[1] = Xim[(c0+1)*16+lo];
      gre = wmma4(mRe,bRe,gre); gre = wmma4(nIm,bIm,gre);
      gim = wmma4(mRe,bIm,gim); gim = wmma4(mIm,bRe,gim);
    }
    #pragma unroll
    for (int r=0;r<8;r++){
      int m = row0 + r + 8*hi;
      gXre[m*16+lo] = gre[r]; gXim[m*16+lo] = gim[r];
    }
  } else if (tid >= 256 && tid < 320){
    int o = tid - 256;
    int k = o >> 4, m = (o >> 2) & 3, s = o & 3;
    const float* hr  = Hre + ((size_t)(k*BX+b)*MXX + m)*NTX;
    const float* hii = Him + ((size_t)(k*BX+b)*MXX + m)*NTX;
    cplx a = cmk(0.f,0.f);
    for (int t=0;t<64;t++)
      a = cadd(a, cmul(cmk(hr[t],hii[t]), cmk(Xre[t*16+k*4+s], Xim[t*16+k*4+s])));
    Are[o] = a.x; Aim[o] = a.y;
  } else if (computeW && tid >= 320 && tid < 384){
    int o = tid - 320;
    int k = o >> 4, m = (o >> 2) & 3, r = o & 3;
    const float* hr  = Hre + ((size_t)(k*BX+b)*MXX + m)*NTX;
    const float* hii = Him + ((size_t)(k*BX+b)*MXX + m)*NTX;
    cplx a = cmk(0.f,0.f);
    for (int t=0;t<64;t++)
      a = cadd(a, cmul(cmk(hr[t],hii[t]), cmk(Fp[(b*256+t*4+r)*2], Fp[(b*256+t*4+r)*2+1])));
    HFre[o] = a.x; HFim[o] = a.y;
  }
  __syncthreads();

  // ---- phase C: gA_k = (2/ln2) * G^-1 A  (4 threads, 4x4 Cholesky inverse) ----
  if (tid < 4){
    cplx G[4][4], L[4][4], Gi[4][4];
    buildG(Are, Aim, tid, G);
    (void)chol4(G, L);
    ginv4(L, Gi);
    for (int m=0;m<4;m++) for (int s=0;s<4;s++){
      cplx acc = cmk(0.f,0.f);
      for (int p=0;p<4;p++)
        acc = cadd(acc, cmul(Gi[m][p], cmk(Are[tid*16+p*4+s], Aim[tid*16+p*4+s])));
      gAre[tid*16+m*4+s] = INV_LN2_X2 * acc.x;
      gAim[tid*16+m*4+s] = INV_LN2_X2 * acc.y;
    }
  }
  __syncthreads();
  // P_k = gA_k W_k^H  (P[k][m][r] = sum_s gA[m,s] conj(W[r,s]))
  if (tid < 64){
    int k = tid >> 4, m = (tid >> 2) & 3, r = tid & 3;
    cplx acc = cmk(0.f,0.f);
    for (int s=0;s<4;s++){
      cplx ga = cmk(gAre[k*16+m*4+s], gAim[k*16+m*4+s]);
      cplx wv = cmk(Wp[(b*64+(k*4+r)*4+s)*2], Wp[(b*64+(k*4+r)*4+s)*2+1]);
      acc = cadd(acc, cmulc(ga, wv));
    }
    Pre[tid] = acc.x; Pim[tid] = acc.y;
  }
  __syncthreads();

  if (!computeW){
    if (tid < 256){ // gFc[t,r] = sum_{k,m} conj(H[k,m,t]) P[k][m,r]
      int t = tid >> 2, r = tid & 3;
      cplx acc = cmk(0.f,0.f);
      for (int k=0;k<4;k++) for (int m=0;m<4;m++){
        cplx h = cmk(Hre[((size_t)(k*BX+b)*MXX+m)*NTX+t], Him[((size_t)(k*BX+b)*MXX+m)*NTX+t]);
        acc = cadd(acc, cmul(cmk(h.x,-h.y), cmk(Pre[(k*4+m)*4+r], Pim[(k*4+m)*4+r])));
      }
      gFc[(b*256+tid)*2] = acc.x; gFc[(b*256+tid)*2+1] = acc.y;
    } else {        // gFr[t,r] = (2/NT^2) sum_{k,m} gX[t,km] conj(W[k,r,m])
      int o = tid - 256, t = o >> 2, r = o & 3;
      cplx acc = cmk(0.f,0.f);
      for (int k=0;k<4;k++) for (int m=0;m<4;m++){
        cplx gx = cmk(gXre[t*16+k*4+m], gXim[t*16+k*4+m]);
        cplx wv = cmk(Wp[(b*64+(k*4+r)*4+m)*2], Wp[(b*64+(k*4+r)*4+m)*2+1]);
        acc = cadd(acc, cmulc(gx, wv));
      }
      gFr[(b*256+o)*2] = TAU_GSCL*acc.x; gFr[(b*256+o)*2+1] = TAU_GSCL*acc.y;
    }
  } else {
    if (tid < 64){  // gWc[k][r][s] = sum_m conj(HF[k][m][r]) gA[k][m][s]
      int k = tid >> 4, r = (tid >> 2) & 3, s = tid & 3;
      cplx acc = cmk(0.f,0.f);
      for (int m=0;m<4;m++){
        cplx hf = cmk(HFre[(k*4+m)*4+r], HFim[(k*4+m)*4+r]);
        cplx ga = cmk(gAre[k*16+m*4+s], gAim[k*16+m*4+s]);
        acc = cadd(acc, cmul(cmk(hf.x,-hf.y), ga));
      }
      gWc[(b*64+tid)*2] = acc.x; gWc[(b*64+tid)*2+1] = acc.y;
    } else if (tid < 128){ // gWr[k][r][m] = (2/NT^2) sum_t conj(F[t,r]) gX[t,km]
      int o = tid - 64;
      int k = o >> 4, r = (o >> 2) & 3, m = o & 3;
      cplx acc = cmk(0.f,0.f);
      for (int t=0;t<64;t++){
        cplx f = cmk(Fp[(b*256+t*4+r)*2], Fp[(b*256+t*4+r)*2+1]);
        cplx gx = cmk(gXre[t*16+k*4+m], gXim[t*16+k*4+m]);
        acc = cadd(acc, cmul(cmk(f.x,-f.y), gx));
      }
      gWr[(b*64+o)*2] = TAU_GSCL*acc.x; gWr[(b*64+o)*2+1] = TAU_GSCL*acc.y;
    }
  }
}

// F += s0*(gFc - omega*gFr) + dq ; F /= |F| ; refresh FW
__global__ __launch_bounds__(256) void k_updF(
    float* Fp, const float* Wp, float* FWp,
    const float* gFc, const float* gFr, const float* dqF,
    const float* step, int outer, int jj)
{
  int b = blockIdx.x, tid = threadIdx.x;
  float s0 = step[outer*(NINX+KUSR)];
  const float* dq = dqF + jj*512;
  int idx = b*256 + tid;
  float fx = Fp[idx*2]   + s0*(gFc[idx*2]   - OMEGAX*gFr[idx*2])   + dq[2*tid];
  float fy = Fp[idx*2+1] + s0*(gFc[idx*2+1] - OMEGAX*gFr[idx*2+1]) + dq[2*tid+1];
  float inv = 1.f/(sqrtf(fx*fx+fy*fy) + 1e-12f);
  Fp[idx*2] = fx*inv; Fp[idx*2+1] = fy*inv;
  __syncthreads();
  for (int i = tid; i < 1024; i += 256){
    int k = i >> 8, t = (i >> 2) & 63, m = i & 3;
    cplx s = cmk(0.f,0.f);
    for (int r=0;r<4;r++){
      cplx f = cmk(Fp[(b*256+t*4+r)*2], Fp[(b*256+t*4+r)*2+1]);
      cplx w = cmk(Wp[(b*64+(k*4+r)*4+m)*2], Wp[(b*64+(k*4+r)*4+m)*2+1]);
      s = cadd(s, cmul(f,w));
    }
    FWp[(b*1024+i)*2] = s.x; FWp[(b*1024+i)*2+1] = s.y;
  }
}

// W += sw*(gWc - omega*gWr) + dqW ; renormalize power ; refresh FW
__global__ __launch_bounds__(256) void k_updW(
    const float* Fp, float* Wp, float* FWp,
    const float* gWc, const float* gWr, const float* dqW,
    const float* step, const int* PtPtr, int outer)
{
  int b = blockIdx.x, tid = threadIdx.x;
  __shared__ float red[256];
  float sw = 0.25f*(step[outer*(NINX+KUSR)+1] + step[outer*(NINX+KUSR)+2]
                  + step[outer*(NINX+KUSR)+3] + step[outer*(NINX+KUSR)+4]);
  if (tid < 64){
    int idx = b*64 + tid;
    Wp[idx*2]   += sw*(gWc[idx*2]   - OMEGAX*gWr[idx*2])   + dqW[2*tid];
    Wp[idx*2+1] += sw*(gWc[idx*2+1] - OMEGAX*gWr[idx*2+1]) + dqW[2*tid+1];
  }
  __syncthreads();
  float acc = 0.f;
  for (int i = tid; i < 1024; i += 256){
    int k = i >> 8, t = (i >> 2) & 63, m = i & 3;
    cplx s = cmk(0.f,0.f);
    for (int r=0;r<4;r++){
      cplx f = cmk(Fp[(b*256+t*4+r)*2], Fp[(b*256+t*4+r)*2+1]);
      cplx w = cmk(Wp[(b*64+(k*4+r)*4+m)*2], Wp[(b*64+(k*4+r)*4+m)*2+1]);
      s = cadd(s, cmul(f,w));
    }
    FWp[(b*1024+i)*2] = s.x; FWp[(b*1024+i)*2+1] = s.y;
    acc += s.x*s.x + s.y*s.y;
  }
  red[tid] = acc; __syncthreads();
  for (int st=128; st; st>>=1){ if (tid<st) red[tid]+=red[tid+st]; __syncthreads(); }
  float scale = sqrtf((float)PtPtr[0] / (red[0] + 1e-12f));
  if (tid < 64){
    int idx = b*64 + tid;
    Wp[idx*2] *= scale; Wp[idx*2+1] *= scale;
  }
  __syncthreads();
  for (int i = tid; i < 1024; i += 256){
    FWp[(b*1024+i)*2] *= scale; FWp[(b*1024+i)*2+1] *= scale;
  }
}

// pack outputs: rates[B,21], taus[B,21], F (complex pairs), W (complex pairs)
__global__ __launch_bounds__(256) void k_final(
    const float* rates, const float* taus, const float* Fp, const float* Wp, float* out)
{
  int i = blockIdx.x*blockDim.x + threadIdx.x;
  const int n1 = BX*(NOUTX+1);
  const int n2 = n1 + BX*(NOUTX+1);
  const int n3 = n2 + BX*256*2;
  const int n4 = n3 + BX*64*2;
  if (i < n1)       out[i] = rates[i];
  else if (i < n2)  out[i] = taus[i-n1];
  else if (i < n3)  out[i] = Fp[i-n2];
  else if (i < n4)  out[i] = Wp[i-n3];
}

// ============================ launcher ============================
extern "C" void kernel_launch(void* const* d_in, const int* in_sizes, int n_in,
                              void* d_out, int out_size, void* d_ws, size_t ws_size,
                              hipStream_t stream)
{
  (void)in_sizes; (void)n_in; (void)out_size; (void)ws_size;
  const float* Hre  = (const float*)d_in[0];
  const float* Him  = (const float*)d_in[1];
  const float* Rre  = (const float*)d_in[2];
  const float* Rim  = (const float*)d_in[3];
  const float* F0re = (const float*)d_in[4];
  const float* F0im = (const float*)d_in[5];
  const float* W0re = (const float*)d_in[6];
  const float* W0im = (const float*)d_in[7];
  const float* step = (const float*)d_in[8];
  const float* FbW1 = (const float*)d_in[9];
  const float* Fbb1 = (const float*)d_in[10];
  const float* FbW2 = (const float*)d_in[11];
  const float* Fbb2 = (const float*)d_in[12];
  const float* Fbqw = (const float*)d_in[13];
  const float* FbW3 = (const float*)d_in[14];
  const float* Fbb3 = (const float*)d_in[15];
  const float* WbW1 = (const float*)d_in[16];
  const float* Wbb1 = (const float*)d_in[17];
  const float* WbW2 = (const float*)d_in[18];
  const float* Wbb2 = (const float*)d_in[19];
  const float* Wbqw = (const float*)d_in[20];
  const float* WbW3 = (const float*)d_in[21];
  const float* Wbb3 = (const float*)d_in[22];
  const float* PW   = (const float*)d_in[23];
  const float* Pb   = (const float*)d_in[24];
  const int*   Pt   = (const int*)  d_in[25];

  float* w = (float*)d_ws;
  float* Fp    = w; w += BX*256*2;
  float* Wp    = w; w += BX*64*2;
  float* FWp   = w; w += BX*1024*2;
  float* gFcP  = w; w += BX*256*2;
  float* gFrP  = w; w += BX*256*2;
  float* gWcP  = w; w += BX*64*2;
  float* gWrP  = w; w += BX*64*2;
  float* vP    = w; w += 80;
  float* dqFP  = w; w += NINX*512;
  float* dqWP  = w; w += 128;
  float* ratesP= w; w += BX*(NOUTX+1);
  float* tausP = w; w += BX*(NOUTX+1);

  k_init<<<BX,256,0,stream>>>(F0re,F0im,W0re,W0im,Pt,Fp,Wp,FWp);
  k_rate_tau<<<BX,512,0,stream>>>(FWp,Hre,Him,Rre,Rim,ratesP,tausP,0);
  for (int o = 0; o < NOUTX; ++o){
    k_state<<<1,512,0,stream>>>(Fp,Wp,Hre,Him,Rre,Rim,PW,Pb,vP);
    k_qblock<<<NINX+1,64,0,stream>>>(vP, FbW1,Fbb1,FbW2,Fbb2,Fbqw,FbW3,Fbb3,
                                     WbW1,Wbb1,WbW2,Wbb2,Wbqw,WbW3,Wbb3, dqFP,dqWP);
    for (int jj = 0; jj < NINX; ++jj){
      k_grads<<<BX,512,0,stream>>>(FWp,Fp,Wp,Hre,Him,Rre,Rim,gFcP,gFrP,gWcP,gWrP,0);
      k_updF<<<BX,256,0,stream>>>(Fp,Wp,FWp,gFcP,gFrP,dqFP,step,o,jj);
      if (jj == 0){
        k_grads<<<BX,512,0,stream>>>(FWp,Fp,Wp,Hre,Him,Rre,Rim,gFcP,gFrP,gWcP,gWrP,1);
        k_updW<<<BX,256,0,stream>>>(Fp,Wp,FWp,gWcP,gWrP,dqWP,step,Pt,o);
      }
    }
    k_rate_tau<<<BX,512,0,stream>>>(FWp,Hre,Him,Rre,Rim,ratesP,tausP,o+1);
  }
  k_final<<<682,256,0,stream>>>(ratesP,tausP,Fp,Wp,(float*)d_out);
}